// NoopSegmenter_21569325761126
// MI455X (gfx1250) — compile-verified
//
#include <hip/hip_runtime.h>
#include <hip/hip_bf16.h>

#define B_ 32
#define T_ 2048
#define D_ 1024
#define S_ 128

typedef __attribute__((ext_vector_type(2))) float v2f;
typedef __attribute__((ext_vector_type(8))) float v8f;

// ---------------------------------------------------------------------------
// Kernel 1: per-batch segment bookkeeping.
//  - seg_id[b,t]     = cumsum(boundary)-1  (boundary[0] forced to 1)
//  - seg_start[b,s]  = first frame with seg_id >= s (T if none), s in [0,128]
//  - inv[b,s]        = 1/count if count>0 else 0
//  - out_rowsum[b,s] = mask.sum(-1) = 1.0 if count>0 else 0.0   (output #2)
//  - out_bound[b,t]  = float(in_boundary[b,t])                  (output #3)
// One block (256 threads) per batch; 8 frames per thread + LDS scan.
// ---------------------------------------------------------------------------
__global__ __launch_bounds__(256) void seg_setup_kernel(
    const int* __restrict__ in_boundary,   // [B, T+1]
    int*       __restrict__ seg_id,        // [B, T]
    int*       __restrict__ seg_start,     // [B, S+1]
    float*     __restrict__ inv,           // [B, S]
    float*     __restrict__ out_rowsum,    // [B, S]
    float*     __restrict__ out_bound)     // [B, T+1]
{
    __shared__ int scan[256];
    __shared__ int ss[S_ + 1];
    __shared__ int counts[S_];

    const int b   = blockIdx.x;
    const int tid = threadIdx.x;
    const int* ib = in_boundary + (long)b * (T_ + 1);

    for (int s = tid; s < S_ + 1; s += 256) ss[s] = T_;
    for (int s = tid; s < S_;     s += 256) counts[s] = 0;

    // per-thread local inclusive sums over 8 consecutive frames
    int bv[8], loc[8];
    const int base = tid * 8;
    int run = 0;
#pragma unroll
    for (int k = 0; k < 8; ++k) {
        const int t = base + k;
        const int v = (t == 0) ? 1 : ib[t];   // force boundary at frame 0
        bv[k] = v;
        run += v;
        loc[k] = run;
    }
    scan[tid] = run;
    __syncthreads();

    // Hillis-Steele inclusive scan over the 256 thread totals
    for (int off = 1; off < 256; off <<= 1) {
        const int v = (tid >= off) ? scan[tid - off] : 0;
        __syncthreads();
        scan[tid] += v;
        __syncthreads();
    }
    const int excl = scan[tid] - run;

#pragma unroll
    for (int k = 0; k < 8; ++k) {
        const int t   = base + k;
        const int sid = excl + loc[k] - 1;
        seg_id[(long)b * T_ + t] = sid;
        if (bv[k] && sid >= 0 && sid <= S_) ss[sid] = t;   // unique writer per sid
        if (sid >= 0 && sid < S_) atomicAdd(&counts[sid], 1);
    }
    __syncthreads();

    for (int s = tid; s < S_; s += 256) {
        const int c = counts[s];
        inv[(long)b * S_ + s]        = (c > 0) ? 1.0f / (float)c : 0.0f;
        out_rowsum[(long)b * S_ + s] = (c > 0) ? 1.0f : 0.0f;
    }
    for (int s = tid; s < S_ + 1; s += 256)
        seg_start[(long)b * (S_ + 1) + s] = ss[s];
    for (int t = tid; t <= T_; t += 256)
        out_bound[(long)b * (T_ + 1) + t] = (float)ib[t];
}

// ---------------------------------------------------------------------------
// Kernel 2: pooled[b] = onehot(seg_id)^T @ x[b], scaled by inv afterwards.
// Grid (D/64, B); 256 threads = 8 wave32s. Wave w owns segment rows
// [16w, 16w+16) and a 16x64 output tile (4 x v8f accumulators). It only
// walks t-chunks of 4 inside [seg_start[16w], seg_start[16w+16]) -- seg_id
// is monotone so that range is contiguous (~3.7x fewer WMMAs than dense).
// A-fragment is a binary one-hot built from seg_id compares (exact f32),
// B-fragment loaded from x per the 16x16x4 f32 layout:
//   lanes 0-15 : K = t0+0, t0+1     lanes 16-31 : K = t0+2, t0+3
// ---------------------------------------------------------------------------
__global__ __launch_bounds__(256) void seg_pool_wmma_kernel(
    const float* __restrict__ x,          // [B, T, D]
    const int*   __restrict__ seg_id,     // [B, T]
    const int*   __restrict__ seg_start,  // [B, S+1]
    const float* __restrict__ inv,        // [B, S]
    float*       __restrict__ pooled)     // [B, S, D]
{
    __shared__ int   lseg[T_ + 4];
    __shared__ int   lstart[S_ + 1];
    __shared__ float linv[S_];

    const int b   = blockIdx.y;
    const int d0  = blockIdx.x * 64;
    const int tid = threadIdx.x;

    for (int t = tid; t < T_; t += 256) lseg[t] = seg_id[(long)b * T_ + t];
    if (tid < 4)        lseg[T_ + tid] = -1;              // sentinel: matches no row
    if (tid < S_ + 1)   lstart[tid] = seg_start[(long)b * (S_ + 1) + tid];
    if (tid < S_)       linv[tid]   = inv[(long)b * S_ + tid];
    __syncthreads();

    const int wave  = tid >> 5;
    const int lane  = tid & 31;
    const int hi    = lane >> 4;      // 0: K=0,1   1: K=2,3
    const int lrow  = lane & 15;
    const int s0    = wave * 16;
    const int koff  = hi * 2;
    const int myrow = s0 + lrow;

    const int tb = lstart[s0] & ~3;                 // wave-uniform
    const int te = lstart[s0 + 16];                 // wave-uniform (<= T)

    const float* xb = x + (long)b * T_ * D_;

    v8f c0 = {}, c1 = {}, c2 = {}, c3 = {};

    for (int t0 = tb; t0 < te; t0 += 4) {
        const int ta = t0 + koff;
        const int sA = lseg[ta];
        const int sB = lseg[ta + 1];

        v2f a;
        a.x = (sA == myrow) ? 1.0f : 0.0f;          // exact binary one-hot
        a.y = (sB == myrow) ? 1.0f : 0.0f;

        // clamp tail reads into bounds; A==0 there so values don't matter
        const int t_0 = (ta     < T_) ? ta     : (T_ - 1);
        const int t_1 = (ta + 1 < T_) ? ta + 1 : (T_ - 1);
        const float* r0 = xb + (long)t_0 * D_ + d0 + lrow;
        const float* r1 = xb + (long)t_1 * D_ + d0 + lrow;

        v2f bf;
        bf.x = r0[0];  bf.y = r1[0];
        c0 = __builtin_amdgcn_wmma_f32_16x16x4_f32(false, a, false, bf, (short)0, c0, false, false);
        bf.x = r0[16]; bf.y = r1[16];
        c1 = __builtin_amdgcn_wmma_f32_16x16x4_f32(false, a, false, bf, (short)0, c1, false, false);
        bf.x = r0[32]; bf.y = r1[32];
        c2 = __builtin_amdgcn_wmma_f32_16x16x4_f32(false, a, false, bf, (short)0, c2, false, false);
        bf.x = r0[48]; bf.y = r1[48];
        c3 = __builtin_amdgcn_wmma_f32_16x16x4_f32(false, a, false, bf, (short)0, c3, false, false);
    }

    // scale by 1/count and store; C/D layout: VGPR r -> row r (+8 for lanes>=16)
    float* ob = pooled + (long)b * S_ * D_;
#pragma unroll
    for (int r = 0; r < 8; ++r) {
        const int  srow   = s0 + r + hi * 8;
        const float iv    = linv[srow];
        const long rowoff = (long)srow * D_ + d0 + lrow;
        ob[rowoff +  0] = c0[r] * iv;
        ob[rowoff + 16] = c1[r] * iv;
        ob[rowoff + 32] = c2[r] * iv;
        ob[rowoff + 48] = c3[r] * iv;
    }
}

// ---------------------------------------------------------------------------
extern "C" void kernel_launch(void* const* d_in, const int* in_sizes, int n_in,
                              void* d_out, int out_size, void* d_ws, size_t ws_size,
                              hipStream_t stream) {
    const float* x  = (const float*)d_in[0];        // [B,T,D] f32
    const int*   ib = (const int*)d_in[1];          // [B,T+1] i32

    float* out    = (float*)d_out;
    float* pooled = out;                             // B*S*D
    float* rowsum = out + (long)B_ * S_ * D_;        // B*S
    float* obound = rowsum + (long)B_ * S_;          // B*(T+1)

    char* ws = (char*)d_ws;
    int*   seg_id    = (int*)ws;                                          // B*T ints
    int*   seg_startp = (int*)(ws + sizeof(int) * (size_t)B_ * T_);       // B*(S+1) ints
    float* invp      = (float*)(ws + sizeof(int) * ((size_t)B_ * T_ + (size_t)B_ * (S_ + 1)));

    seg_setup_kernel<<<B_, 256, 0, stream>>>(ib, seg_id, seg_startp, invp, rowsum, obound);

    dim3 grid(D_ / 64, B_);
    seg_pool_wmma_kernel<<<grid, 256, 0, stream>>>(x, seg_id, seg_startp, invp, pooled);
}